// DA_GNN_63471026700634
// MI455X (gfx1250) — compile-verified
//
#include <hip/hip_runtime.h>
#include <math.h>

#define NN    200000
#define EE    6400000
#define DD    3
#define EDD   4
#define GG    128
#define NCC   2
#define EPSBN 1e-5f
#define FLTMAX 3.402823466e+38f

// ---- workspace layout (float offsets) ----
#define WS_ST      0        // S_node[3], T_node[3]
#define WS_AB      8        // A_e[4], B_e[4]
#define WS_LAYER   16       // 5 * 32: P[9] @+0, Q[9] @+9, S[3] @+18, T[3] @+21
#define WS_PART    256      // 256 blocks * 8 partial fields
#define WS_POOLED  8448     // 128*6
#define WS_H       16384    // h [N*3]
#define WS_T       620544   // agg / t [N*3]

typedef float v2f __attribute__((ext_vector_type(2)));
typedef float v8f __attribute__((ext_vector_type(8)));

#if defined(__has_builtin)
#  if __has_builtin(__builtin_amdgcn_wmma_f32_16x16x4_f32)
#    define USE_WMMA_F32X4 1
#  endif
#endif

struct ConvPtrs { const float* nnw[5]; const float* nnb[5]; };
struct HeadPtrs { const float* W[8]; const float* B[8]; const float* Gm[6]; const float* Be[6]; };

// ---------------- helpers ----------------
__device__ __forceinline__ float bsum256(float v, float* red) {
  int tid = threadIdx.x;
  red[tid] = v; __syncthreads();
  for (int s = 128; s > 0; s >>= 1) { if (tid < s) red[tid] += red[tid + s]; __syncthreads(); }
  float r = red[0]; __syncthreads(); return r;
}
__device__ __forceinline__ float bmax256(float v, float* red) {
  int tid = threadIdx.x;
  red[tid] = v; __syncthreads();
  for (int s = 128; s > 0; s >>= 1) { if (tid < s) red[tid] = fmaxf(red[tid], red[tid + s]); __syncthreads(); }
  float r = red[0]; __syncthreads(); return r;
}
__device__ __forceinline__ int lbound(const int* a, int n, int v) {
  int lo = 0, hi = n;
  while (lo < hi) { int m = (lo + hi) >> 1; if (a[m] < v) lo = m + 1; else hi = m; }
  return lo;
}

// ---------------- stage 1: batch stats of z = x@nw + nb ----------------
__global__ void k_xstats(const float* __restrict__ x, const float* __restrict__ nw,
                         const float* __restrict__ nb, float* __restrict__ part) {
  __shared__ float red[256];
  __shared__ float w[12];
  int tid = threadIdx.x;
  if (tid < 12) w[tid] = (tid < 9) ? nw[tid] : nb[tid - 9];
  __syncthreads();
  float s[3] = {0,0,0}, ss[3] = {0,0,0};
  for (int n = blockIdx.x * 256 + tid; n < NN; n += 256 * 256) {
    float x0 = x[3*n], x1 = x[3*n+1], x2 = x[3*n+2];
#pragma unroll
    for (int j = 0; j < 3; ++j) {
      float z = fmaf(x0, w[j], fmaf(x1, w[3+j], fmaf(x2, w[6+j], w[9+j])));
      s[j] += z; ss[j] += z * z;
    }
  }
#pragma unroll
  for (int q = 0; q < 6; ++q) {
    float r = bsum256(q < 3 ? s[q] : ss[q-3], red);
    if (tid == 0) part[blockIdx.x * 8 + q] = r;
  }
}

// stats of edge_attr scalar -> fields 6,7
__global__ void k_estats(const float* __restrict__ ea, float* __restrict__ part) {
  __shared__ float red[256];
  int tid = threadIdx.x;
  float s = 0.f, ss = 0.f;
  for (int e = blockIdx.x * 256 + tid; e < EE; e += 256 * 256) {
    float a = ea[e]; s += a; ss += a * a;
  }
  float r0 = bsum256(s, red);
  float r1 = bsum256(ss, red);
  if (tid == 0) { part[blockIdx.x * 8 + 6] = r0; part[blockIdx.x * 8 + 7] = r1; }
}

// ---------------- stage 2: finalize stats, build all fused constants ----------------
__global__ void k_finalize(float* __restrict__ ws,
                           const float* __restrict__ ng, const float* __restrict__ nbe,
                           const float* __restrict__ ew, const float* __restrict__ eb,
                           const float* __restrict__ eg, const float* __restrict__ ebe,
                           ConvPtrs cp) {
  __shared__ float red[256];
  __shared__ float tot[8];
  __shared__ float AB[8];
  int tid = threadIdx.x;
  float v[8];
#pragma unroll
  for (int q = 0; q < 8; ++q) v[q] = ws[WS_PART + tid * 8 + q];
#pragma unroll
  for (int q = 0; q < 8; ++q) {
    float r = bsum256(v[q], red);
    if (tid == 0) tot[q] = r;
  }
  __syncthreads();
  if (tid == 0) {
    for (int j = 0; j < 3; ++j) {
      float m = tot[j] / (float)NN;
      float var = tot[3+j] / (float)NN - m * m;
      float S = rsqrtf(var + EPSBN) * ng[j];
      ws[WS_ST + j] = S;
      ws[WS_ST + 3 + j] = nbe[j] - m * S;
    }
    float ma = tot[6] / (float)EE;
    float va = tot[7] / (float)EE - ma * ma;
    for (int j = 0; j < 4; ++j) {
      float me = fmaf(ma, ew[j], eb[j]);
      float ve = va * ew[j] * ew[j];
      float rs = rsqrtf(ve + EPSBN);
      float A = ew[j] * rs * eg[j];
      float B = (eb[j] - me) * rs * eg[j] + ebe[j];
      ws[WS_AB + j] = A; ws[WS_AB + 4 + j] = B;
      AB[j] = A; AB[4 + j] = B;
    }
  }
  __syncthreads();
  // P[l] = A . nnw ; Q[l] = B . nnw + nnb   (We = a*P + Q per edge)
  if (tid < 45) {
    int l = tid / 9, idx = tid % 9;
    const float* nnw = cp.nnw[l];
    float P = 0.f, Q = 0.f;
#pragma unroll
    for (int j = 0; j < 4; ++j) {
      P = fmaf(AB[j],     nnw[j * 9 + idx], P);
      Q = fmaf(AB[4 + j], nnw[j * 9 + idx], Q);
    }
    Q += cp.nnb[l][idx];
    ws[WS_LAYER + 32 * l + idx]     = P;
    ws[WS_LAYER + 32 * l + 9 + idx] = Q;
  }
}

// ---------------- h0 = BN(x@nw+nb) applied ----------------
__global__ void k_h0(const float* __restrict__ x, const float* __restrict__ nw,
                     const float* __restrict__ nb, const float* __restrict__ ST,
                     float* __restrict__ h) {
  __shared__ float w[12], st[6];
  int tid = threadIdx.x;
  if (tid < 12) w[tid] = (tid < 9) ? nw[tid] : nb[tid - 9];
  if (tid < 6) st[tid] = ST[tid];
  __syncthreads();
  int n = blockIdx.x * 256 + tid;
  if (n >= NN) return;
  float x0 = x[3*n], x1 = x[3*n+1], x2 = x[3*n+2];
#pragma unroll
  for (int j = 0; j < 3; ++j) {
    float z = fmaf(x0, w[j], fmaf(x1, w[3+j], fmaf(x2, w[6+j], w[9+j])));
    h[3*n + j] = fmaf(z, st[j], st[3 + j]);
  }
}

__global__ void k_clear(float* __restrict__ t, int n) {
  int i = blockIdx.x * 256 + threadIdx.x;
  if (i < n) t[i] = 0.f;
}

// ---------------- edge scatter: msg = a*(h@P) + h@Q, atomic add to agg[dst] ----------------
__global__ void k_edge(const float* __restrict__ ea, const int* __restrict__ src,
                       const int* __restrict__ dst, const float* __restrict__ h,
                       float* __restrict__ agg, const float* __restrict__ PQ) {
  __shared__ float pq[18];
  if (threadIdx.x < 18) pq[threadIdx.x] = PQ[threadIdx.x];
  __syncthreads();
  int e = blockIdx.x * 256 + threadIdx.x;
  if (e >= EE) return;
  if (e + 8192 < EE) __builtin_prefetch(&ea[e + 8192], 0, 0);
  float a = ea[e];
  int s = src[e], d = dst[e];
  float h0 = h[3*s], h1 = h[3*s+1], h2 = h[3*s+2];
#pragma unroll
  for (int o = 0; o < 3; ++o) {
    float p = fmaf(h0, pq[o],     fmaf(h1, pq[3+o],     h2 * pq[6+o]));
    float q = fmaf(h0, pq[9+o],   fmaf(h1, pq[12+o],    h2 * pq[15+o]));
    atomicAdd(&agg[3*d + o], fmaf(a, p, q));
  }
}

// ---------------- node update: t = relu(agg + h@root + bias), accumulate stats ----------------
__global__ void k_node_stats(const float* __restrict__ h, float* __restrict__ t,
                             const float* __restrict__ root, const float* __restrict__ bias,
                             float* __restrict__ part) {
  __shared__ float red[256];
  __shared__ float rb[12];
  int tid = threadIdx.x;
  if (tid < 12) rb[tid] = (tid < 9) ? root[tid] : bias[tid - 9];
  __syncthreads();
  float s[3] = {0,0,0}, ss[3] = {0,0,0};
  for (int n = blockIdx.x * 256 + tid; n < NN; n += 256 * 256) {
    float h0 = h[3*n], h1 = h[3*n+1], h2 = h[3*n+2];
#pragma unroll
    for (int o = 0; o < 3; ++o) {
      float v = t[3*n + o] + fmaf(h0, rb[o], fmaf(h1, rb[3+o], fmaf(h2, rb[6+o], rb[9+o])));
      v = fmaxf(v, 0.f);
      t[3*n + o] = v;
      s[o] += v; ss[o] += v * v;
    }
  }
#pragma unroll
  for (int q = 0; q < 6; ++q) {
    float r = bsum256(q < 3 ? s[q] : ss[q-3], red);
    if (tid == 0) part[blockIdx.x * 8 + q] = r;
  }
}

__global__ void k_node_fin(float* __restrict__ ws, const float* __restrict__ g,
                           const float* __restrict__ be, int l) {
  __shared__ float red[256];
  int tid = threadIdx.x;
  float v[6];
#pragma unroll
  for (int q = 0; q < 6; ++q) v[q] = ws[WS_PART + tid * 8 + q];
  float tot[6];
#pragma unroll
  for (int q = 0; q < 6; ++q) tot[q] = bsum256(v[q], red);
  if (tid == 0) {
    for (int j = 0; j < 3; ++j) {
      float m = tot[j] / (float)NN;
      float var = tot[3+j] / (float)NN - m * m;
      float S = rsqrtf(var + EPSBN) * g[j];
      ws[WS_LAYER + 32 * l + 18 + j] = S;
      ws[WS_LAYER + 32 * l + 21 + j] = be[j] - m * S;
    }
  }
}

__global__ void k_apply(const float* __restrict__ t, float* __restrict__ h,
                        const float* __restrict__ ST) {
  __shared__ float st[6];
  int tid = threadIdx.x;
  if (tid < 6) st[tid] = ST[tid];
  __syncthreads();
  int n = blockIdx.x * 256 + tid;
  if (n >= NN) return;
#pragma unroll
  for (int o = 0; o < 3; ++o) h[3*n + o] = fmaf(t[3*n + o], st[o], st[3 + o]);
}

// ---------------- pooling: per-graph mean + max (batch_vec sorted) ----------------
__global__ void k_pool(const float* __restrict__ h, const int* __restrict__ batch,
                       float* __restrict__ pooled) {
  __shared__ float red[256];
  int g = blockIdx.x, tid = threadIdx.x;
  int lo = lbound(batch, NN, g);
  int hi = lbound(batch, NN, g + 1);
  float s[3] = {0,0,0}, mx[3] = {-FLTMAX, -FLTMAX, -FLTMAX};
  for (int n = lo + tid; n < hi; n += 256) {
#pragma unroll
    for (int j = 0; j < 3; ++j) { float v = h[3*n + j]; s[j] += v; mx[j] = fmaxf(mx[j], v); }
  }
  float sums[3], maxs[3];
#pragma unroll
  for (int j = 0; j < 3; ++j) sums[j] = bsum256(s[j], red);
#pragma unroll
  for (int j = 0; j < 3; ++j) maxs[j] = bmax256(mx[j], red);
  if (tid == 0) {
    int cnt = hi - lo;
    float c = (float)(cnt > 0 ? cnt : 1);
    for (int j = 0; j < 3; ++j) {
      pooled[g * 6 + j] = sums[j] / c;
      pooled[g * 6 + 3 + j] = (cnt > 0) ? maxs[j] : 0.f;
    }
  }
}

// ---------------- MLP heads with WMMA f32 16x16x4 ----------------
__global__ void k_head(const float* __restrict__ pooled, HeadPtrs hp, float* __restrict__ out) {
  __shared__ float bufA[128 * 16];
  __shared__ float bufB[128 * 16];
  __shared__ float Wl[16 * 16];
  __shared__ float bl[16], Sc[16], Tc[16];
  const int KinArr[8]  = {6, 12, 9, 6,  6, 16, 8, 4};
  const int NoutArr[8] = {12, 9, 6, 1, 16,  8, 4, 2};
  int tid = threadIdx.x;
  int wv = tid >> 5, lane = tid & 31;
  int col = lane & 15;
  int kb = (lane >= 16) ? 2 : 0;
  int rowA = wv * 16 + (lane & 15);

  for (int head = 0; head < 2; ++head) {
    // load pooled [128,6] into padded [128,16]
    for (int idx = tid; idx < 128 * 16; idx += 256) {
      int c = idx & 15, r = idx >> 4;
      bufA[idx] = (c < 6) ? pooled[r * 6 + c] : 0.f;
    }
    __syncthreads();
    float* A = bufA; float* Bb = bufB;

    for (int li = 0; li < 3; ++li) {
      int L = head * 4 + li;
      int K = KinArr[L], Nout = NoutArr[L];
      const float* W = hp.W[L];
      // pad weights to 16x16 (zeros outside [K,Nout]); bias to 16
      for (int idx = tid; idx < 256; idx += 256) {
        int k = idx >> 4, n = idx & 15;
        Wl[idx] = (k < K && n < Nout) ? W[k * Nout + n] : 0.f;
      }
      if (tid < 16) bl[tid] = (tid < NoutArr[L]) ? hp.B[L][tid] : 0.f;
      __syncthreads();

      float accv[8];
#ifdef USE_WMMA_F32X4
      v8f acc = {0.f,0.f,0.f,0.f,0.f,0.f,0.f,0.f};
#pragma unroll
      for (int kc = 0; kc < 4; ++kc) {
        int k0 = kc * 4 + kb;
        v2f a; a.x = A[rowA * 16 + k0]; a.y = A[rowA * 16 + k0 + 1];
        v2f b; b.x = Wl[k0 * 16 + col]; b.y = Wl[(k0 + 1) * 16 + col];
        acc = __builtin_amdgcn_wmma_f32_16x16x4_f32(false, a, false, b, (short)0, acc,
                                                    false, false);
      }
#pragma unroll
      for (int v = 0; v < 8; ++v) accv[v] = acc[v];
#else
#pragma unroll
      for (int v = 0; v < 8; ++v) {
        int ro = wv * 16 + v + ((lane >= 16) ? 8 : 0);
        float sum = 0.f;
        for (int k = 0; k < 16; ++k) sum = fmaf(A[ro * 16 + k], Wl[k * 16 + col], sum);
        accv[v] = sum;
      }
#endif
      // C layout: VGPR v -> row M = v (lanes 0-15) / v+8 (lanes 16-31), N = lane&15
#pragma unroll
      for (int v = 0; v < 8; ++v) {
        int ro = wv * 16 + v + ((lane >= 16) ? 8 : 0);
        float val = fmaxf(accv[v] + bl[col], 0.f);   // bias + relu (pad cols stay 0)
        Bb[ro * 16 + col] = val;
      }
      __syncthreads();
      // BN over the 128-row batch, per column
      int gi = head * 3 + li;
      if (tid < 16) {
        if (tid < Nout) {
          float s = 0.f, ss = 0.f;
          for (int r = 0; r < 128; ++r) { float xv = Bb[r * 16 + tid]; s += xv; ss += xv * xv; }
          float m = s / 128.f, var = ss / 128.f - m * m;
          float sc = rsqrtf(var + EPSBN) * hp.Gm[gi][tid];
          Sc[tid] = sc; Tc[tid] = hp.Be[gi][tid] - m * sc;
        } else { Sc[tid] = 0.f; Tc[tid] = 0.f; }
      }
      __syncthreads();
      for (int idx = tid; idx < 128 * 16; idx += 256) {
        int c = idx & 15;
        Bb[idx] = fmaf(Bb[idx], Sc[c], Tc[c]);
      }
      __syncthreads();
      float* tmp = A; A = Bb; Bb = tmp;
    }
    // final plain linear
    int L = head * 4 + 3;
    int K = KinArr[L], Nout = NoutArr[L];
    const float* W = hp.W[L]; const float* bias = hp.B[L];
    if (tid < 128) {
      for (int c = 0; c < Nout; ++c) {
        float s = bias[c];
        for (int k = 0; k < K; ++k) s = fmaf(A[tid * 16 + k], W[k * Nout + c], s);
        if (head == 0) out[tid] = s;
        else out[128 + tid * NCC + c] = s;
      }
    }
    __syncthreads();
  }
}

// ---------------- host launch ----------------
extern "C" void kernel_launch(void* const* d_in, const int* in_sizes, int n_in,
                              void* d_out, int out_size, void* d_ws, size_t ws_size,
                              hipStream_t stream) {
  (void)in_sizes; (void)n_in; (void)out_size; (void)ws_size;
  float* ws = (float*)d_ws;
  const float* x   = (const float*)d_in[0];
  const float* ea  = (const float*)d_in[1];
  const float* nw  = (const float*)d_in[2];
  const float* nb  = (const float*)d_in[3];
  const float* ng  = (const float*)d_in[4];
  const float* nbe = (const float*)d_in[5];
  const float* ew  = (const float*)d_in[6];
  const float* eb  = (const float*)d_in[7];
  const float* eg  = (const float*)d_in[8];
  const float* ebe = (const float*)d_in[9];
  const int* eidx  = (const int*)d_in[68];
  const int* src   = eidx;
  const int* dst   = eidx + EE;
  const int* batch = (const int*)d_in[69];

  ConvPtrs cp;
  for (int l = 0; l < 5; ++l) {
    cp.nnw[l] = (const float*)d_in[10 + 6 * l + 0];
    cp.nnb[l] = (const float*)d_in[10 + 6 * l + 1];
  }
  HeadPtrs hp;
  for (int i = 0; i < 4; ++i) {  // lp: d_in[40..53]
    int b = 40 + (i < 3 ? i * 4 : 12);
    hp.W[i] = (const float*)d_in[b]; hp.B[i] = (const float*)d_in[b + 1];
    if (i < 3) { hp.Gm[i] = (const float*)d_in[b + 2]; hp.Be[i] = (const float*)d_in[b + 3]; }
  }
  for (int i = 0; i < 4; ++i) {  // dom: d_in[54..67]
    int b = 54 + (i < 3 ? i * 4 : 12);
    hp.W[4 + i] = (const float*)d_in[b]; hp.B[4 + i] = (const float*)d_in[b + 1];
    if (i < 3) { hp.Gm[3 + i] = (const float*)d_in[b + 2]; hp.Be[3 + i] = (const float*)d_in[b + 3]; }
  }

  float* h = ws + WS_H;
  float* t = ws + WS_T;
  int nbN  = (NN + 255) / 256;
  int nbN3 = (NN * 3 + 255) / 256;
  int nbE  = (EE + 255) / 256;

  k_xstats<<<256, 256, 0, stream>>>(x, nw, nb, ws + WS_PART);
  k_estats<<<256, 256, 0, stream>>>(ea, ws + WS_PART);
  k_finalize<<<1, 256, 0, stream>>>(ws, ng, nbe, ew, eb, eg, ebe, cp);
  k_h0<<<nbN, 256, 0, stream>>>(x, nw, nb, ws + WS_ST, h);

  for (int l = 0; l < 5; ++l) {
    const float* root = (const float*)d_in[10 + 6 * l + 2];
    const float* bias = (const float*)d_in[10 + 6 * l + 3];
    const float* g    = (const float*)d_in[10 + 6 * l + 4];
    const float* be   = (const float*)d_in[10 + 6 * l + 5];
    k_clear<<<nbN3, 256, 0, stream>>>(t, NN * 3);
    k_edge<<<nbE, 256, 0, stream>>>(ea, src, dst, h, t, ws + WS_LAYER + 32 * l);
    k_node_stats<<<256, 256, 0, stream>>>(h, t, root, bias, ws + WS_PART);
    k_node_fin<<<1, 256, 0, stream>>>(ws, g, be, l);
    k_apply<<<nbN, 256, 0, stream>>>(t, h, ws + WS_LAYER + 32 * l + 18);
  }

  k_pool<<<GG, 256, 0, stream>>>(h, batch, ws + WS_POOLED);
  k_head<<<1, 256, 0, stream>>>(ws + WS_POOLED, hp, (float*)d_out);
}